// TripletLoss_53102975648041
// MI455X (gfx1250) — compile-verified
//
#include <hip/hip_runtime.h>
#include <hip/hip_bf16.h>

typedef __attribute__((ext_vector_type(2))) float v2f;
typedef __attribute__((ext_vector_type(8))) float v8f;

#define N_PTS 8192
#define DIM   128
#define MARGIN_C 0.3f
#define LDS_STRIDE 132                 // 128 + 4 pad -> conflict-free b64 frag reads
#define TILE_FLOATS (16 * LDS_STRIDE)  // one 16-col x 128-k B tile

// -------- per-call prep: row squared norms + reset mining accumulators ------
__global__ void tl_prep(const float* __restrict__ x, float* __restrict__ sq,
                        unsigned* __restrict__ ap, unsigned* __restrict__ an) {
    int i = blockIdx.x * 256 + threadIdx.x;
    if (i >= N_PTS) return;
    ap[i] = 0u;
    an[i] = 0x7F800000u;               // +inf
    const float4* p = (const float4*)(x + (size_t)i * DIM);
    float s = 0.f;
#pragma unroll
    for (int c = 0; c < DIM / 4; ++c) {
        float4 v = p[c];
        s += v.x * v.x + v.y * v.y + v.z * v.z + v.w * v.w;
    }
    sq[i] = s;
}

// -------- main: X@X^T via fp32 WMMA, async double-buffered B staging --------
// Block = 256 threads (8 waves). Wave w -> rows [blockIdx.x*128 + w*16, +16).
// blockIdx.y selects a 1024-column group (64 tiles of 16 columns).
__launch_bounds__(256)
__global__ void tl_dist(const float* __restrict__ x,
                        const int*   __restrict__ tgt,
                        const float* __restrict__ sq,
                        float*       __restrict__ dist,
                        unsigned*    __restrict__ ap_g,
                        unsigned*    __restrict__ an_g) {
    __shared__ float ldsB[2][TILE_FLOATS];

    const int lane = threadIdx.x & 31;
    const int wave = threadIdx.x >> 5;            // 0..7
    const int half = lane >> 4;                   // 0|1 -> K-pair select / row half
    const int l15  = lane & 15;
    const int rowbase  = blockIdx.x * 128 + wave * 16;
    const int colstart = blockIdx.y * 1024;

    // A fragments: 16 rows x K=128 fp32, 16x16x4 layout
    // lane L holds row (L&15); half-wave picks K offset {0,2}; v2f = K,K+1
    v2f a[32];
    {
        const float* arow = x + (size_t)(rowbase + l15) * DIM + half * 2;
#pragma unroll
        for (int s = 0; s < 32; ++s) a[s] = *(const v2f*)(arow + s * 4);
    }

    // per-lane row metadata (C layout: VGPR r <-> row r + half*8)
    int   rt[8];  float sqm[8], ap[8], an[8];
#pragma unroll
    for (int r = 0; r < 8; ++r) {
        int m  = rowbase + half * 8 + r;
        rt[r]  = tgt[m];
        sqm[r] = sq[m];
        ap[r]  = 0.0f;                 // dist >= 0; diagonal guarantees a positive
        an[r]  = INFINITY;
    }

    // async staging: each thread moves two 16B chunks per tile:
    // chunk (n0, k0) and (n0+8, k0); n = output column in tile, k0 in floats
    const int n0 = threadIdx.x >> 5;              // 0..7
    const int k0 = (threadIdx.x & 31) * 4;        // 0,4,...,124
    const unsigned ldsoff0 =
        (unsigned)(uintptr_t)&ldsB[0][n0 * LDS_STRIDE + k0];        // 16B aligned
    const unsigned ldsoff1 =
        (unsigned)(uintptr_t)&ldsB[0][(n0 + 8) * LDS_STRIDE + k0];
    const unsigned bufstride = (unsigned)(TILE_FLOATS * sizeof(float)); // 8448

    auto stage = [&](int colbase, int buf) {
        unsigned v0 = (unsigned)(((colbase + n0) * DIM + k0) * 4);
        unsigned v1 = v0 + (unsigned)(8 * DIM * 4);
        unsigned l0 = ldsoff0 + (unsigned)buf * bufstride;
        unsigned l1 = ldsoff1 + (unsigned)buf * bufstride;
        asm volatile("global_load_async_to_lds_b128 %0, %1, %2"
                     :: "v"(l0), "v"(v0), "s"(x) : "memory");
        asm volatile("global_load_async_to_lds_b128 %0, %1, %2"
                     :: "v"(l1), "v"(v1), "s"(x) : "memory");
    };

    stage(colstart, 0);                            // prologue: tile 0 -> buf 0

    for (int ct = 0; ct < 64; ++ct) {
        const int buf     = ct & 1;
        const int colbase = colstart + ct * 16;

        if (ct < 63) {
            stage(colbase + 16, buf ^ 1);          // prefetch next tile
            // wait for all but my 2 just-issued loads (async completes in order)
            asm volatile("s_wait_asynccnt 0x2" ::: "memory");
        } else {
            asm volatile("s_wait_asynccnt 0x0" ::: "memory");
        }
        __syncthreads();                           // current tile visible to all waves

        v8f acc = {};
        const float* bb = &ldsB[buf][l15 * LDS_STRIDE + half * 2];
#pragma unroll
        for (int s = 0; s < 32; ++s) {
            v2f b = *(const v2f*)(bb + s * 4);
            acc = __builtin_amdgcn_wmma_f32_16x16x4_f32(
                false, a[s], false, b, (short)0, acc, false, false);
        }

        const int   cn  = colbase + l15;
        const int   tn  = tgt[cn];
        const float sqn = sq[cn];
#pragma unroll
        for (int r = 0; r < 8; ++r) {
            int   m  = rowbase + half * 8 + r;
            float d2 = sqm[r] + sqn - 2.0f * acc[r];
            d2 = fminf(fmaxf(d2, 1e-12f), 1e12f);
            float d  = sqrtf(d2);
            dist[(size_t)m * N_PTS + cn] = d;
            if (rt[r] == tn) ap[r] = fmaxf(ap[r], d);
            else             an[r] = fminf(an[r], d);
        }
        __syncthreads();   // all waves done reading buf before it is restaged
    }

    // reduce across the 16 lanes sharing each row, then one atomic per row
#pragma unroll
    for (int r = 0; r < 8; ++r) {
        float av = ap[r], nv = an[r];
#pragma unroll
        for (int m = 1; m < 16; m <<= 1) {
            av = fmaxf(av, __shfl_xor(av, m, 16));
            nv = fminf(nv, __shfl_xor(nv, m, 16));
        }
        if (l15 == 0) {
            int row = rowbase + half * 8 + r;
            atomicMax(&ap_g[row], __float_as_uint(av));  // uint order == float order (>=0)
            atomicMin(&an_g[row], __float_as_uint(nv));
        }
    }
}

// -------- loss / precision reduction ---------------------------------------
__global__ void tl_finalize(const unsigned* __restrict__ ap_g,
                            const unsigned* __restrict__ an_g,
                            float* __restrict__ out) {
    __shared__ float sl[256], sp[256];
    float accl = 0.f, accp = 0.f;
    for (int i = threadIdx.x; i < N_PTS; i += 256) {
        float ap = __uint_as_float(ap_g[i]);   // * MARGIN2 (=1.0)
        float an = __uint_as_float(an_g[i]);   // / MARGIN3 (=1.0)
        float l  = ap - an + MARGIN_C;
        accl += (l > 0.f) ? l : 0.f;
        accp += (an > ap) ? 1.f : 0.f;
    }
    sl[threadIdx.x] = accl; sp[threadIdx.x] = accp;
    __syncthreads();
    for (int s = 128; s > 0; s >>= 1) {
        if (threadIdx.x < s) {
            sl[threadIdx.x] += sl[threadIdx.x + s];
            sp[threadIdx.x] += sp[threadIdx.x + s];
        }
        __syncthreads();
    }
    if (threadIdx.x == 0) {
        out[0] = sl[0] / (float)N_PTS;
        out[1] = sp[0] / (float)N_PTS;
    }
}

extern "C" void kernel_launch(void* const* d_in, const int* in_sizes, int n_in,
                              void* d_out, int out_size, void* d_ws, size_t ws_size,
                              hipStream_t stream) {
    (void)in_sizes; (void)n_in; (void)out_size; (void)ws_size;
    const float* x   = (const float*)d_in[0];
    const int*   tgt = (const int*)d_in[1];
    float* out = (float*)d_out;

    float*    sq = (float*)d_ws;
    unsigned* ap = (unsigned*)((char*)d_ws + (size_t)N_PTS * 4);
    unsigned* an = (unsigned*)((char*)d_ws + (size_t)N_PTS * 8);
    float* dist = out + 2;   // output order: loss, prec, dist

    tl_prep<<<N_PTS / 256, 256, 0, stream>>>(x, sq, ap, an);
    dim3 grid(N_PTS / 128, 8);
    tl_dist<<<grid, 256, 0, stream>>>(x, tgt, sq, dist, ap, an);
    tl_finalize<<<1, 256, 0, stream>>>(ap, an, out);
}